// PPGNN_39977555591297
// MI455X (gfx1250) — compile-verified
//
#include <hip/hip_runtime.h>
#include <hip/hip_bf16.h>

#define NN     50000
#define EE     800000
#define CIN    128
#define HH     64
#define NC     40
#define LL     15
#define DTc    0.1f
#define JACN   2
#define LNEPS  1e-5f

typedef float v2f __attribute__((ext_vector_type(2)));
typedef float v8f __attribute__((ext_vector_type(8)));
typedef long long i64;

// ---------------- setup kernels ----------------

__global__ void deg_kernel(const i64* __restrict__ ei, float* __restrict__ deg) {
    int e = blockIdx.x * blockDim.x + threadIdx.x;
    if (e < EE) atomicAdd(&deg[(int)ei[EE + e]], 1.0f);
}

__global__ void ew_kernel(const i64* __restrict__ ei, const float* __restrict__ deg,
                          float* __restrict__ ew) {
    int e = blockIdx.x * blockDim.x + threadIdx.x;
    if (e < EE) {
        int s = (int)ei[e], d = (int)ei[EE + e];
        float is = rsqrtf(fmaxf(deg[s], 1.0f));
        float id = rsqrtf(fmaxf(deg[d], 1.0f));
        ew[e] = is * id;
    }
}

__global__ void ones_kernel(float* __restrict__ h) {
    int i = blockIdx.x * blockDim.x + threadIdx.x;     // N*16 quads
    if (i < NN * 16) {
        int r = i >> 4, q = i & 15;
        float4 one = {1.0f, 1.0f, 1.0f, 1.0f};
        ((float4*)(h + (size_t)r * 128 + 64))[q] = one;
    }
}

// ---------------- lift GEMM: h[:, :64] = tanh(x @ lift_w^T + b), f32 WMMA ----------------

__global__ void lift_gemm(const float* __restrict__ x, const float* __restrict__ w,
                          const float* __restrict__ b, float* __restrict__ h) {
    int tile = blockIdx.x;            // 3125 * 4 tiles
    int tm = tile >> 2, tn = tile & 3;
    int lane = threadIdx.x;
    int half = lane >> 4, l16 = lane & 15;
    int arow = tm * 16 + l16;         // A: M = lane%16
    int bcol = tn * 16 + l16;         // B: N = lane%16
    v8f acc = {0.f,0.f,0.f,0.f,0.f,0.f,0.f,0.f};
    for (int k = 0; k < CIN; k += 4) {
        int kk = k + 2 * half;        // lanes 0-15: K=k,k+1 ; lanes 16-31: K=k+2,k+3
        v2f a, bb;
        a.x  = x[arow * CIN + kk];
        a.y  = x[arow * CIN + kk + 1];
        bb.x = w[bcol * CIN + kk];    // B[k][n] = lift_w[n][k]
        bb.y = w[bcol * CIN + kk + 1];
        acc = __builtin_amdgcn_wmma_f32_16x16x4_f32(false, a, false, bb,
                                                    (short)0, acc, false, false);
    }
    float bias = b[bcol];
#pragma unroll
    for (int v = 0; v < 8; ++v) {
        int m = v + 8 * half;         // C/D layout: M = v + 8*(lane/16), N = lane%16
        h[(tm * 16 + m) * 128 + tn * 16 + l16] = tanhf(acc[v] + bias);
    }
}

// ---------------- per-layer elementwise: rhs + initial Jacobi guess (float4) ----------------

__global__ void rhs_kernel(const float* __restrict__ h,
                           const float* __restrict__ alpha, const float* __restrict__ beta,
                           const float* __restrict__ dxp, const float* __restrict__ dyp,
                           int l, float* __restrict__ rhs, float* __restrict__ Zb) {
    int i = blockIdx.x * blockDim.x + threadIdx.x;     // N*32 channel-quads
    if (i >= NN * 32) return;
    int r = i >> 5, q = i & 31;
    bool isX = q < 16;
    float a = alpha[l], bt = beta[l];
    float d = isX ? dxp[l] : dyp[l];
    float inv = 1.0f / (1.0f + DTc * d);
    const float4* hp = (const float4*)(h + (size_t)r * 128);
    float4 Xv = hp[isX ? q : q - 16];
    float4 Yv = hp[isX ? q + 16 : q];
    const float* xa = (const float*)&Xv;
    const float* ya = (const float*)&Yv;
    float4 rv, zv;
    float* ra = (float*)&rv;
    float* za = (float*)&zv;
#pragma unroll
    for (int j = 0; j < 4; ++j) {
        float XY = xa[j] * ya[j];
        float rr = isX ? (xa[j] + DTc * (a * xa[j] - bt * XY))
                       : (ya[j] + DTc * (bt * XY - a * ya[j]));
        ra[j] = rr;
        za[j] = rr * inv;
    }
    ((float4*)rhs)[i] = rv;
    ((float4*)Zb)[i]  = zv;
}

// ---------------- SpMM: one wave32 per edge, b128 gather, scalar index base ----------------

__global__ void spmm_kernel(const i64* __restrict__ ei, const float* __restrict__ ew,
                            const float* __restrict__ Zb, float* __restrict__ tmp) {
    int t = blockIdx.x * blockDim.x + threadIdx.x;     // E*32 threads
    int e = t >> 5;                                    // wave-uniform edge id
    int lane = t & 31;
    if (e >= EE) return;                               // uniform branch (exact grid)
    int s = __builtin_amdgcn_readfirstlane((int)ei[e]);
    int d = __builtin_amdgcn_readfirstlane((int)ei[EE + e]);
    float wgt = ew[e];
    float4 z = ((const float4*)(Zb + (size_t)s * 128))[lane];
    float* tp = tmp + (size_t)d * 128 + lane * 4;
    atomicAdd(tp + 0, wgt * z.x);
    atomicAdd(tp + 1, wgt * z.y);
    atomicAdd(tp + 2, wgt * z.z);
    atomicAdd(tp + 3, wgt * z.w);
}

__global__ void jacobi_update(const float* __restrict__ rhs, const float* __restrict__ tmp,
                              const float* __restrict__ dxp, const float* __restrict__ dyp,
                              int l, float* __restrict__ Zb) {
    int i = blockIdx.x * blockDim.x + threadIdx.x;     // N*32 channel-quads
    if (i >= NN * 32) return;
    int q = i & 31;
    float d = (q < 16) ? dxp[l] : dyp[l];
    float s = DTc * d;
    float inv = 1.0f / (1.0f + s);
    float4 rv = ((const float4*)rhs)[i];
    float4 tv = ((const float4*)tmp)[i];
    float4 zv;
    zv.x = (rv.x + s * tv.x) * inv;
    zv.y = (rv.y + s * tv.y) * inv;
    zv.z = (rv.z + s * tv.z) * inv;
    zv.w = (rv.w + s * tv.w) * inv;
    ((float4*)Zb)[i] = zv;
}

// ---------------- gate + LayerNorm: one wave32 per node, float4 per lane ----------------

__global__ void gate_ln(float* __restrict__ h, const float* __restrict__ Zb,
                        const float* __restrict__ tau,
                        const float* __restrict__ lnw, const float* __restrict__ lnb, int l) {
    int wid  = (blockIdx.x * blockDim.x + threadIdx.x) >> 5;
    int lane = threadIdx.x & 31;
    if (wid >= NN) return;
    float g = 1.0f / (1.0f + expf(-tau[l]));
    float4* hp = (float4*)(h + (size_t)wid * 128);
    const float4* zp = (const float4*)(Zb + (size_t)wid * 128);
    float4 hv = hp[lane];
    float4 zv = zp[lane];
    float nv[4];
    float* ha = (float*)&hv;
    float* za = (float*)&zv;
    float s = 0.f, ss = 0.f;
#pragma unroll
    for (int j = 0; j < 4; ++j) {
        nv[j] = (1.0f - g) * ha[j] + g * za[j];
        s += nv[j]; ss += nv[j] * nv[j];
    }
#pragma unroll
    for (int off = 16; off > 0; off >>= 1) {
        s  += __shfl_xor(s,  off, 32);
        ss += __shfl_xor(ss, off, 32);
    }
    float mu  = s * (1.0f / 128.0f);
    float var = ss * (1.0f / 128.0f) - mu * mu;
    float inv = rsqrtf(var + LNEPS);
    float4 lw = ((const float4*)(lnw + (size_t)l * 128))[lane];
    float4 lb = ((const float4*)(lnb + (size_t)l * 128))[lane];
    const float* lwa = (const float*)&lw;
    const float* lba = (const float*)&lb;
    float4 outv;
    float* oa = (float*)&outv;
#pragma unroll
    for (int j = 0; j < 4; ++j)
        oa[j] = (nv[j] - mu) * inv * lwa[j] + lba[j];
    hp[lane] = outv;
}

// ---------------- output GEMM: out = ls * X @ out_w^T + out_b, f32 WMMA, padded to 48 ----------------

__global__ void out_gemm(const float* __restrict__ h, const float* __restrict__ w,
                         const float* __restrict__ b, const float* __restrict__ ls,
                         float* __restrict__ out) {
    int tile = blockIdx.x;            // 3125 * 3 tiles
    int tm = tile / 3, tn = tile % 3;
    int lane = threadIdx.x;
    int half = lane >> 4, l16 = lane & 15;
    int arow = tm * 16 + l16;
    int bcol = tn * 16 + l16;
    v8f acc = {0.f,0.f,0.f,0.f,0.f,0.f,0.f,0.f};
    for (int k = 0; k < HH; k += 4) {
        int kk = k + 2 * half;
        v2f a, bb;
        a.x  = h[arow * 128 + kk];            // X = h[:, :64]
        a.y  = h[arow * 128 + kk + 1];
        bb.x = (bcol < NC) ? w[bcol * HH + kk]     : 0.0f;
        bb.y = (bcol < NC) ? w[bcol * HH + kk + 1] : 0.0f;
        acc = __builtin_amdgcn_wmma_f32_16x16x4_f32(false, a, false, bb,
                                                    (short)0, acc, false, false);
    }
    float scale = ls[0];
    int n = tn * 16 + l16;
    if (n < NC) {
        float bias = b[n];
#pragma unroll
        for (int v = 0; v < 8; ++v) {
            int m = v + 8 * half;
            out[(tm * 16 + m) * NC + n] = scale * acc[v] + bias;
        }
    }
}

// ---------------- host orchestration ----------------

extern "C" void kernel_launch(void* const* d_in, const int* in_sizes, int n_in,
                              void* d_out, int out_size, void* d_ws, size_t ws_size,
                              hipStream_t stream) {
    (void)in_sizes; (void)n_in; (void)out_size; (void)ws_size;
    const float* x      = (const float*)d_in[0];
    const i64*   ei     = (const i64*)  d_in[1];
    const float* lift_w = (const float*)d_in[2];
    const float* lift_b = (const float*)d_in[3];
    const float* alpha  = (const float*)d_in[4];
    const float* beta   = (const float*)d_in[5];
    const float* dxp    = (const float*)d_in[6];
    const float* dyp    = (const float*)d_in[7];
    const float* tau    = (const float*)d_in[8];
    const float* lnw    = (const float*)d_in[9];
    const float* lnb    = (const float*)d_in[10];
    const float* out_w  = (const float*)d_in[11];
    const float* out_b  = (const float*)d_in[12];
    const float* lscale = (const float*)d_in[13];
    float* out = (float*)d_out;

    float* ws  = (float*)d_ws;
    float* h   = ws;                          // N*128
    float* rhs = h   + (size_t)NN * 128;      // N*128
    float* Zb  = rhs + (size_t)NN * 128;      // N*128
    float* tmp = Zb  + (size_t)NN * 128;      // N*128
    float* ew  = tmp + (size_t)NN * 128;      // E
    float* deg = ew  + (size_t)EE;            // N

    hipMemsetAsync(deg, 0, NN * sizeof(float), stream);
    deg_kernel<<<(EE + 255) / 256, 256, 0, stream>>>(ei, deg);
    ew_kernel <<<(EE + 255) / 256, 256, 0, stream>>>(ei, deg, ew);
    ones_kernel<<<(NN * 16 + 255) / 256, 256, 0, stream>>>(h);
    lift_gemm<<<3125 * 4, 32, 0, stream>>>(x, lift_w, lift_b, h);

    for (int l = 0; l < LL; ++l) {
        rhs_kernel<<<(NN * 32 + 255) / 256, 256, 0, stream>>>(h, alpha, beta, dxp, dyp,
                                                              l, rhs, Zb);
        for (int j = 0; j < JACN; ++j) {
            hipMemsetAsync(tmp, 0, (size_t)NN * 128 * sizeof(float), stream);
            spmm_kernel<<<(EE * 32) / 256, 256, 0, stream>>>(ei, ew, Zb, tmp);
            jacobi_update<<<(NN * 32 + 255) / 256, 256, 0, stream>>>(rhs, tmp, dxp, dyp,
                                                                     l, Zb);
        }
        gate_ln<<<(NN * 32 + 255) / 256, 256, 0, stream>>>(h, Zb, tau, lnw, lnb, l);
    }

    out_gemm<<<3125 * 3, 32, 0, stream>>>(h, out_w, out_b, lscale, out);
}